// Embed2GraphByLearnableAttention_53420803228022
// MI455X (gfx1250) — compile-verified
//
#include <hip/hip_runtime.h>
#include <hip/hip_bf16.h>
#include <math.h>

typedef __attribute__((ext_vector_type(2))) float v2f;
typedef __attribute__((ext_vector_type(8))) float v8f;

static constexpr int Bc = 4;
static constexpr int Nc = 4096;
static constexpr int Dc = 128;
static constexpr int ROWS_PER_BLOCK = 16;

// ---------------------------------------------------------------------------
// Kernel 1: s = x @ [a1, a2] via V_WMMA_F32_16X16X4_F32 (f32 matrix pipe).
// One wave handles a 16-row tile of the flattened (B*N, 128) x matrix.
// B operand (4x16): col0 = a1 chunk, col1 = a2 chunk, cols 2..15 = 0.
// C tile (16x16): col0 = s_i for the 16 rows, col1 = s_j.
// ---------------------------------------------------------------------------
__global__ void gemv_wmma_f32(const float* __restrict__ x,
                              const float* __restrict__ a,
                              float* __restrict__ si,
                              float* __restrict__ sj) {
    const int lane  = threadIdx.x & 31;
    const int wave  = threadIdx.x >> 5;
    const int tile  = blockIdx.x * (blockDim.x >> 5) + wave;   // 0..(B*N/16 - 1)
    const int row_base = tile * 16;

    const int m     = lane & 15;      // A: row  M;  B: col N;  C: col N
    const int khalf = lane >> 4;      // K-half select per ISA 16x4 layout

    const float* __restrict__ xrow = x + (size_t)(row_base + m) * Dc;

    v8f c = {};
    #pragma unroll
    for (int k0 = 0; k0 < Dc; k0 += 4) {
        const int kA = k0 + 2 * khalf;      // VGPR0 K, VGPR1 K = kA + 1
        v2f av, bv;
        av.x = xrow[kA];
        av.y = xrow[kA + 1];
        float b0 = 0.f, b1 = 0.f;
        if (m == 0)      { b0 = a[kA];       b1 = a[kA + 1]; }        // a1
        else if (m == 1) { b0 = a[Dc + kA];  b1 = a[Dc + kA + 1]; }   // a2
        bv.x = b0; bv.y = b1;
        // 8-arg form: (neg_a, A, neg_b, B, c_mod, C, reuse_a, reuse_b)
        c = __builtin_amdgcn_wmma_f32_16x16x4_f32(false, av, false, bv,
                                                  (short)0, c, false, false);
    }

    // C layout: lane L, VGPR r -> row = r + 8*(L>>4), col = L&15.
    if (m < 2) {
        float* __restrict__ dst = (m == 0) ? si : sj;
        const int rbase = row_base + 8 * khalf;
        #pragma unroll
        for (int r = 0; r < 8; ++r) dst[rbase + r] = c[r];
    }
}

// ---------------------------------------------------------------------------
// Block reductions over 256 threads (8 wave32s).
// ---------------------------------------------------------------------------
__device__ __forceinline__ float block_max(float v, float* red) {
    const int lane = threadIdx.x & 31, wave = threadIdx.x >> 5;
    #pragma unroll
    for (int off = 16; off > 0; off >>= 1) v = fmaxf(v, __shfl_xor(v, off, 32));
    if (lane == 0) red[wave] = v;
    __syncthreads();
    float r = red[0];
    #pragma unroll
    for (int w = 1; w < 8; ++w) r = fmaxf(r, red[w]);
    __syncthreads();
    return r;
}

__device__ __forceinline__ float block_sum(float v, float* red) {
    const int lane = threadIdx.x & 31, wave = threadIdx.x >> 5;
    #pragma unroll
    for (int off = 16; off > 0; off >>= 1) v += __shfl_xor(v, off, 32);
    if (lane == 0) red[wave] = v;
    __syncthreads();
    float r = red[0];
    #pragma unroll
    for (int w = 1; w < 8; ++w) r += red[w];
    __syncthreads();
    return r;
}

// ---------------------------------------------------------------------------
// Kernel 2: for each row (b,i): softmax_j( leakyrelu(s_i[b,i] + s_j[b,j]) ).
// 256 threads/block, 16 rows/block. s_j[b,:] staged once in LDS (16 KB).
// Output stores are nontemporal (256 MB stream, never re-read).
// ---------------------------------------------------------------------------
__global__ void softmax_rows(const float* __restrict__ si,
                             const float* __restrict__ sj,
                             float* __restrict__ out) {
    __shared__ float sjl[Nc];
    __shared__ float red[8];

    const int blocks_per_b = Nc / ROWS_PER_BLOCK;          // 256
    const int b     = blockIdx.x / blocks_per_b;
    const int chunk = blockIdx.x % blocks_per_b;

    const float* __restrict__ sjb = sj + (size_t)b * Nc;
    for (int j = threadIdx.x; j < Nc; j += 256) sjl[j] = sjb[j];
    __syncthreads();

    for (int rr = 0; rr < ROWS_PER_BLOCK; ++rr) {
        const int i   = chunk * ROWS_PER_BLOCK + rr;
        const float svi = si[(size_t)b * Nc + i];

        float e[16];
        float m = -INFINITY;
        #pragma unroll
        for (int t = 0; t < 16; ++t) {
            const float v0 = svi + sjl[threadIdx.x + t * 256];
            const float v  = v0 > 0.f ? v0 : 0.2f * v0;
            e[t] = v;
            m = fmaxf(m, v);
        }
        m = block_max(m, red);

        float s = 0.f;
        #pragma unroll
        for (int t = 0; t < 16; ++t) {
            const float v = __expf(e[t] - m);
            e[t] = v;
            s += v;
        }
        s = block_sum(s, red);
        const float inv = 1.f / s;

        float* __restrict__ orow = out + ((size_t)b * Nc + i) * Nc;
        #pragma unroll
        for (int t = 0; t < 16; ++t) {
            __builtin_nontemporal_store(e[t] * inv, &orow[threadIdx.x + t * 256]);
        }
        __syncthreads();   // red[] reused next row
    }
}

// ---------------------------------------------------------------------------
extern "C" void kernel_launch(void* const* d_in, const int* in_sizes, int n_in,
                              void* d_out, int out_size, void* d_ws, size_t ws_size,
                              hipStream_t stream) {
    const float* x = (const float*)d_in[0];   // (B, N, D) f32
    const float* a = (const float*)d_in[1];   // (2D, 1)  f32
    float* out = (float*)d_out;               // (B, N, N, 1) f32

    float* si = (float*)d_ws;                 // B*N floats
    float* sj = si + Bc * Nc;                 // B*N floats (128 KB total)

    // Kernel 1: B*N/16 = 1024 wave-tiles, 8 waves (256 thr) per block.
    const int tiles = (Bc * Nc) / 16;
    gemv_wmma_f32<<<tiles / 8, 256, 0, stream>>>(x, a, si, sj);

    // Kernel 2: one block per 16 rows.
    const int blocks = Bc * (Nc / ROWS_PER_BLOCK);          // 1024
    softmax_rows<<<blocks, 256, 0, stream>>>(si, sj, out);
}